// ARBlock_15530601742818
// MI455X (gfx1250) — compile-verified
//
#include <hip/hip_runtime.h>

#define LATENT  512
#define HIDDEN  1024
#define GATES   3072            // 3*HIDDEN
#define BATCH   64
#define SEQ     512
#define M_TOTAL (BATCH * SEQ)   // 32768 rows of x_proj

typedef __attribute__((ext_vector_type(2))) float v2f;
typedef __attribute__((ext_vector_type(4))) float v4f;
typedef __attribute__((ext_vector_type(8))) float v8f;
typedef int          v4i __attribute__((vector_size(16)));
typedef unsigned int v4u __attribute__((vector_size(16)));
typedef int          v8i __attribute__((vector_size(32)));

typedef __attribute__((address_space(3))) void lvoid;   // LDS pointee type

#if defined(__has_builtin)
#if __has_builtin(__builtin_amdgcn_global_load_async_to_lds_b128)
#define USE_ASYNC_LDS 1
#endif
#if __has_builtin(__builtin_amdgcn_tensor_load_to_lds) && \
    __has_builtin(__builtin_amdgcn_s_wait_tensorcnt)
#define USE_TDM 1
#endif
#endif

__device__ __forceinline__ float fast_sigmoid(float x) {
    return 1.0f / (1.0f + __expf(-x));
}

__device__ __forceinline__ v8f wmma_f32(v2f a, v2f b, v8f c) {
    // V_WMMA_F32_16X16X4_F32 : D = A(16x4) * B(4x16) + C(16x16), all f32
    return __builtin_amdgcn_wmma_f32_16x16x4_f32(
        false, a, false, b, (short)0, c, false, false);
}

// 16-byte global -> LDS stage (async DMA when available).
__device__ __forceinline__ void stage16(const float* __restrict__ g, float* l) {
#ifdef USE_ASYNC_LDS
    __builtin_amdgcn_global_load_async_to_lds_b128((v4i*)g, (v4i*)l, 0, 0);
#else
    *(v4f*)l = *(const v4f*)g;
#endif
}

__device__ __forceinline__ void async_wait_all() {
#ifdef USE_ASYNC_LDS
    asm volatile("s_wait_asynccnt 0x0" ::: "memory");
#endif
}

__device__ __forceinline__ void tdm_wait_all() {
#ifdef USE_TDM
    __builtin_amdgcn_s_wait_tensorcnt(0);
#endif
}

// Byte offset of a generic pointer within the wave's LDS segment.
__device__ __forceinline__ unsigned lds_byte_offset(void* p) {
    return (unsigned)(unsigned long long)(lvoid*)p;
}

__global__ void zero_kernel(float* __restrict__ p, int n) {
    int i = blockIdx.x * blockDim.x + threadIdx.x;
    if (i < n) p[i] = 0.0f;
}

// -----------------------------------------------------------------------------
// Phase 1:  X[m, g] = sum_i Z[m, i] * W_ih[g, i] + b_ih[g]
// Register-blocked: each wave computes a 16x64 strip (4 accumulators) so the
// A fragment is loaded once per 4 WMMAs.
// -----------------------------------------------------------------------------
__global__ __launch_bounds__(256) void xproj_kernel(
    const float* __restrict__ Z,
    const float* __restrict__ Wih,
    const float* __restrict__ bih,
    float* __restrict__ X)
{
    const int lane = threadIdx.x & 31;
    const int wave = threadIdx.x >> 5;
    const int job  = blockIdx.x * 8 + wave;       // 2048 * 48 = 98304 jobs
    const int mT   = job / (GATES / 64);
    const int gS   = job % (GATES / 64);
    const int m0   = mT * 16;
    const int g0   = gS * 64;
    const int hi   = lane >> 4;
    const int lr   = lane & 15;

    const float* arow = Z + (size_t)(m0 + lr) * LATENT;
    const float* bp0  = Wih + (size_t)(g0 +  0 + lr) * LATENT;
    const float* bp1  = Wih + (size_t)(g0 + 16 + lr) * LATENT;
    const float* bp2  = Wih + (size_t)(g0 + 32 + lr) * LATENT;
    const float* bp3  = Wih + (size_t)(g0 + 48 + lr) * LATENT;

    v8f c0, c1, c2, c3;
    {
        const float s0 = bih[g0 +  0 + lr];
        const float s1 = bih[g0 + 16 + lr];
        const float s2 = bih[g0 + 32 + lr];
        const float s3 = bih[g0 + 48 + lr];
#pragma unroll
        for (int i = 0; i < 8; ++i) { c0[i] = s0; c1[i] = s1; c2[i] = s2; c3[i] = s3; }
    }

#pragma unroll 4
    for (int k0 = 0; k0 < LATENT; k0 += 4) {
        const int ko = k0 + 2 * hi;
        v2f a  = *(const v2f*)(arow + ko);        // shared A fragment
        v2f b0 = *(const v2f*)(bp0 + ko);
        v2f b1 = *(const v2f*)(bp1 + ko);
        v2f b2 = *(const v2f*)(bp2 + ko);
        v2f b3 = *(const v2f*)(bp3 + ko);
        c0 = wmma_f32(a, b0, c0);
        c1 = wmma_f32(a, b1, c1);
        c2 = wmma_f32(a, b2, c2);
        c3 = wmma_f32(a, b3, c3);
    }

    // C/D layout: VGPR r -> M = r + 8*hi, N = lr
    float* ob = X + (size_t)m0 * GATES + g0 + lr;
#pragma unroll
    for (int r = 0; r < 8; ++r) {
        const size_t ro = (size_t)(r + 8 * hi) * GATES;
        ob[ro +  0] = c0[r];
        ob[ro + 16] = c1[r];
        ob[ro + 32] = c2[r];
        ob[ro + 48] = c3[r];
    }
}

// -----------------------------------------------------------------------------
// Phase 2 staging: per K-chunk (KC=64) a block needs
//   h slab : 64 rows x 64 f32            -> async global->LDS (all threads)
//   W slab : 3 gates x 32 rows x 64 f32  -> one TDM descriptor (wave 0)
// LDS rows are padded to 76 floats (TDM pad_enable: 12 DWORDs per 64) so the
// v2f fragment reads below are bank-conflict free.
// -----------------------------------------------------------------------------
#define KC        64
#define LSTRIDE   76                    // floats; 304B rows: 16B aligned
#define ROWS_H    64
#define ROWS_W    96                    // 3 gates * 32 j-columns
#define ROWS_T    (ROWS_H + ROWS_W)     // 160
#define BUF_FL    (ROWS_T * LSTRIDE)    // 12160 floats / buffer
#define SMEM_BYTES (2 * BUF_FL * 4)     // 97280 bytes
#define NCHUNK    (HIDDEN / KC)         // 16

__device__ __forceinline__ void stage_h_chunk(
    float* __restrict__ dstbuf, const float* __restrict__ hprev,
    int kc, int tid)
{
    // 64 rows x 16 chunks of 16B = 1024 transfers; 4 per thread.
#pragma unroll
    for (int i = 0; i < 4; ++i) {
        const int idx = tid + i * 256;
        const int row = idx >> 4;
        const int c4  = (idx & 15) << 2;
        stage16(hprev + (size_t)row * HIDDEN + kc + c4,
                dstbuf + row * LSTRIDE + c4);
    }
}

#ifdef USE_TDM
// One 3D TDM descriptor: tile (x=KC, y=32 rows, z=3 gates), element 4B,
// dim0_stride = HIDDEN, dim1_stride = HIDDEN*HIDDEN (gate plane), LDS padding
// 12 DWORDs after every 64 DWORDs (-> LSTRIDE=76 rows).
__device__ __forceinline__ void issue_w_tdm(
    const float* __restrict__ Whh, int jblk0, int kc, unsigned lds_byte)
{
    const unsigned long long ga =
        (unsigned long long)(const void*)(Whh + (size_t)jblk0 * HIDDEN + kc);

    v4u g0;
    g0[0] = 1u;                                             // count=1 (valid)
    g0[1] = lds_byte;                                       // lds_addr (bytes)
    g0[2] = (unsigned)(ga & 0xFFFFFFFFu);                   // global_addr[31:0]
    g0[3] = (unsigned)((ga >> 32) & 0x1FFFFFFu) | (2u << 30); // addr[56:32]|type=2

    v8i g1;
    g1[0] = (int)((2u << 16)        // data_size = 4B
                | (1u << 20)        // pad_enable
                | (5u << 22)        // pad_interval: 2^(5+1)=64 DWORDs
                | (11u << 25));     // pad_amount : 11+1 =12 DWORDs
    g1[1] = (int)(64u << 16);       // tensor_dim0[15:0]=64  (abar addr = 0)
    g1[2] = (int)(32u << 16);       // tensor_dim0 hi=0 | tensor_dim1[15:0]=32
    g1[3] = (int)(64u << 16);       // tensor_dim1 hi=0 | tile_dim0=64
    g1[4] = (int)(32u | (3u << 16));// tile_dim1=32 | tile_dim2=3
    g1[5] = (int)HIDDEN;            // tensor_dim0_stride lo32 = 1024
    g1[6] = 0;                      // stride0 hi16 | stride1[15:0] (2^20 -> 0)
    g1[7] = (int)(((size_t)HIDDEN * HIDDEN) >> 16);  // stride1[47:16] = 16

    v4i g2;
    g2[0] = 3;                      // tensor_dim2 = 3 gate planes
    g2[1] = 0; g2[2] = 0; g2[3] = 0;
    v4i g3 = {0, 0, 0, 0};
    v8i g4 = {0, 0, 0, 0, 0, 0, 0, 0};   // extra group (6-arg toolchain form)

    __builtin_amdgcn_tensor_load_to_lds(g0, g1, g2, g3, g4, 0);
}
#else
__device__ __forceinline__ void stage_w_chunk(
    float* __restrict__ dstbuf, const float* __restrict__ Whh,
    int jblk0, int kc, int tid)
{
    // 96 rows x 16 chunks of 16B = 1536 transfers; 6 per thread.
#pragma unroll
    for (int i = 0; i < 6; ++i) {
        const int idx  = tid + i * 256;
        const int wrow = idx >> 4;           // 0..95
        const int c4   = (idx & 15) << 2;
        const int gate = wrow >> 5;
        const int rr   = wrow & 31;
        stage16(Whh + ((size_t)gate * HIDDEN + jblk0 + rr) * HIDDEN + kc + c4,
                dstbuf + wrow * LSTRIDE + c4);
    }
}
#endif

__device__ __forceinline__ void stage_all(
    float* __restrict__ buf, const float* __restrict__ hprev,
    const float* __restrict__ Whh, int jblk0, int kc, int tid, int wave)
{
    stage_h_chunk(buf, hprev, kc, tid);
#ifdef USE_TDM
    if (wave == 0)
        issue_w_tdm(Whh, jblk0, kc, lds_byte_offset(buf + ROWS_H * LSTRIDE));
#else
    (void)wave;
    stage_w_chunk(buf + ROWS_H * LSTRIDE, Whh, jblk0, kc, tid);
#endif
}

__global__ __launch_bounds__(256) void gru_step_kernel(
    const float* __restrict__ hprev,
    float* __restrict__ hnew,
    const float* __restrict__ X,
    const float* __restrict__ Whh,
    const float* __restrict__ bhh,
    float* __restrict__ out,
    int t)
{
    extern __shared__ float smem[];

    const int tid  = threadIdx.x;
    const int lane = tid & 31;
    const int wave = tid >> 5;
    const int hi   = lane >> 4;
    const int lr   = lane & 15;

    const int jblk0 = blockIdx.x * 32;     // 32 j-columns per block
    const int btile = wave & 3;            // 4 batch tiles (full B=64)
    const int jj    = wave >> 2;           // 2 j-subtiles
    const int b0    = btile * 16;
    const int j     = jblk0 + jj * 16 + lr;

    // Accumulators initialized with b_hh (gh = h @ W_hh^T + b_hh)
    v8f cr, cz, cn;
    {
        const float br = bhh[j];
        const float bz = bhh[HIDDEN + j];
        const float bn = bhh[2 * HIDDEN + j];
#pragma unroll
        for (int i = 0; i < 8; ++i) { cr[i] = br; cz[i] = bz; cn[i] = bn; }
    }

    // Prologue: stage chunk 0
    stage_all(smem, hprev, Whh, jblk0, 0, tid, wave);
    async_wait_all();
    tdm_wait_all();
    __syncthreads();

    for (int c = 0; c < NCHUNK; ++c) {
        if (c + 1 < NCHUNK)
            stage_all(smem + ((c + 1) & 1) * BUF_FL, hprev, Whh, jblk0,
                      (c + 1) * KC, tid, wave);

        const float* buf = smem + (c & 1) * BUF_FL;
        const float* ah  = buf + (b0 + lr) * LSTRIDE;                      // h rows
        const float* w0  = buf + (ROWS_H +  0 + jj * 16 + lr) * LSTRIDE;   // gate r
        const float* w1  = buf + (ROWS_H + 32 + jj * 16 + lr) * LSTRIDE;   // gate z
        const float* w2  = buf + (ROWS_H + 64 + jj * 16 + lr) * LSTRIDE;   // gate n

#pragma unroll 4
        for (int k0 = 0; k0 < KC; k0 += 4) {
            const int ko = k0 + 2 * hi;
            v2f a  = *(const v2f*)(ah + ko);       // shared A fragment (ds_load)
            v2f b1 = *(const v2f*)(w0 + ko);
            v2f b2 = *(const v2f*)(w1 + ko);
            v2f b3 = *(const v2f*)(w2 + ko);
            cr = wmma_f32(a, b1, cr);
            cz = wmma_f32(a, b2, cz);
            cn = wmma_f32(a, b3, cn);
        }

        async_wait_all();      // prefetched h slab (chunk c+1) has landed
        tdm_wait_all();        // wave 0: W slab TDM for chunk c+1 complete
        __syncthreads();       // publish to all waves; buffers safe to swap
    }

    // Fused gates + state update + output store.
#pragma unroll
    for (int r = 0; r < 8; ++r) {
        const int bb = b0 + r + 8 * hi;
        const float* xrow = X + ((size_t)bb * SEQ + t) * GATES;
        const float xr = xrow[j];
        const float xz = xrow[HIDDEN + j];
        const float xn = xrow[2 * HIDDEN + j];
        const float hp = hprev[(size_t)bb * HIDDEN + j];

        const float rg = fast_sigmoid(xr + cr[r]);
        const float ug = fast_sigmoid(xz + cz[r]);
        const float ng = tanhf(xn + rg * cn[r]);
        const float hv = (1.0f - ug) * ng + ug * hp;

        hnew[(size_t)bb * HIDDEN + j]            = hv;
        out[((size_t)bb * SEQ + t) * HIDDEN + j] = hv;
    }
}

extern "C" void kernel_launch(void* const* d_in, const int* in_sizes, int n_in,
                              void* d_out, int out_size, void* d_ws, size_t ws_size,
                              hipStream_t stream) {
    const float* z   = (const float*)d_in[0];   // (64, 512, 512)
    const float* Wih = (const float*)d_in[1];   // (3072, 512)
    const float* Whh = (const float*)d_in[2];   // (3072, 1024)
    const float* bih = (const float*)d_in[3];   // (3072,)
    const float* bhh = (const float*)d_in[4];   // (3072,)
    float* out = (float*)d_out;                 // (64, 512, 1024)

    float* X  = (float*)d_ws;                       // 32768*3072 f32
    float* h0 = X + (size_t)M_TOTAL * GATES;        // 64*1024 f32 (ping)
    float* h1 = h0 + (size_t)BATCH * HIDDEN;        // 64*1024 f32 (pong)

    // h(-1) = 0
    zero_kernel<<<(BATCH * HIDDEN + 255) / 256, 256, 0, stream>>>(h0, BATCH * HIDDEN);

    // Phase 1: input projection GEMM (16x64 strip per wave, 8 waves/block)
    const int jobs = (M_TOTAL / 16) * (GATES / 64);   // 98304
    xproj_kernel<<<jobs / 8, 256, 0, stream>>>(z, Wih, bih, X);

    // Phase 2: sequential scan; 32 blocks cover (B=64) x (H=1024) per step
    for (int t = 0; t < SEQ; ++t) {
        const float* hp = (t & 1) ? h1 : h0;
        float*       hn = (t & 1) ? h0 : h1;
        gru_step_kernel<<<HIDDEN / 32, 256, SMEM_BYTES, stream>>>(
            hp, hn, X, Whh, bhh, out, t);
    }
}